// UniNetModel_31576599560597
// MI455X (gfx1250) — compile-verified
//
#include <hip/hip_runtime.h>
#include <math.h>

// ---------------------------------------------------------------------------
// UniNet fused multi-scale upsample (WMMA f32 GEMM) + gaussian blur + max
//
//   maps(256x256)  = Wy(256x112) @ Tcat(112x256)        <- V_WMMA_F32_16X16X4_F32
//   Tcat           = [X0 @ Rx0^T ; X1 @ Rx1^T ; X2 @ Rx2^T] (2-tap horiz, VALU)
//   scores[b]      = max( gauss5x5_reflect(maps[b]) )
//
// d_out layout: scores[256] then maps[256*256*256] (reference return order).
// ---------------------------------------------------------------------------

typedef __attribute__((ext_vector_type(2))) float v2f;
typedef __attribute__((ext_vector_type(8))) float v8f;

#define OUT_HW     256
#define K_TOT      112          // 64 + 32 + 16 rows after horizontal pass
#define WY_STRIDE  113          // odd row stride -> conflict-free A-frag loads
#define TC_STRIDE  264          // 2-row delta = 16 banks -> disjoint B half-waves
#define WY_FLOATS  (OUT_HW * WY_STRIDE)        // 28,928
#define TC_FLOATS  (K_TOT * TC_STRIDE)         // 29,568
#define X0_OFF     (WY_FLOATS + TC_FLOATS)     // 58,496
#define X1_OFF     (X0_OFF + 64 * 64)          // 62,592
#define X2_OFF     (X1_OFF + 32 * 32)          // 63,616
#define SMEM_FLOATS (X2_OFF + 16 * 16)         // 63,872 floats = 255,488 B (<320KB/WGP)

__global__ void uninet_upsample_gemm(const float* __restrict__ in0,
                                     const float* __restrict__ in1,
                                     const float* __restrict__ in2,
                                     float* __restrict__ maps) {
  extern __shared__ float smem[];
  float* WY = smem;               // [256][113] vertical interp weights
  float* TC = smem + WY_FLOATS;   // [112][264] horizontally-interpolated rows
  float* X0 = smem + X0_OFF;
  float* X1 = smem + X1_OFF;
  float* X2 = smem + X2_OFF;

  const int tid = threadIdx.x;    // 256 threads = 8 wave32
  const int b   = blockIdx.x;

  // ---- stage inputs into LDS (coalesced) ----
  const float* g0 = in0 + (size_t)b * 4096;
  for (int i = tid; i < 4096; i += 256) X0[i] = g0[i];
  const float* g1 = in1 + (size_t)b * 1024;
  for (int i = tid; i < 1024; i += 256) X1[i] = g1[i];
  const float* g2 = in2 + (size_t)b * 256;
  X2[tid] = g2[tid];

  // ---- zero Wy, then build it (thread t owns output row t) ----
  for (int i = tid; i < WY_FLOATS; i += 256) WY[i] = 0.0f;
  __syncthreads();
  {
    const int i = tid;
    const int offs[3] = {0, 64, 96};
    const int hs[3]   = {64, 32, 16};
#pragma unroll
    for (int s = 0; s < 3; ++s) {
      const int h = hs[s];
      float y = (float)i * (float)(h - 1) / 255.0f;   // linspace(0, h-1, 256)
      int y0 = (int)y; if (y0 > h - 1) y0 = h - 1;
      float fy = y - (float)y0;
      int y1 = y0 + 1; if (y1 > h - 1) y1 = h - 1;
      WY[i * WY_STRIDE + offs[s] + y0] += 1.0f - fy;  // accumulate handles y1==y0
      WY[i * WY_STRIDE + offs[s] + y1] += fy;
    }
  }

  // ---- horizontal 2-tap pass: thread t owns output column j = t ----
  {
    const int j = tid;
    float x = (float)j * 63.0f / 255.0f;
    int   x0 = (int)x;  float fx = x - (float)x0;
    int   x1 = (x0 + 1 > 63) ? 63 : x0 + 1;
    for (int r = 0; r < 64; ++r)
      TC[r * TC_STRIDE + j] = X0[r * 64 + x0] * (1.0f - fx) + X0[r * 64 + x1] * fx;

    x = (float)j * 31.0f / 255.0f;
    x0 = (int)x;  fx = x - (float)x0;
    x1 = (x0 + 1 > 31) ? 31 : x0 + 1;
    for (int r = 0; r < 32; ++r)
      TC[(64 + r) * TC_STRIDE + j] = X1[r * 32 + x0] * (1.0f - fx) + X1[r * 32 + x1] * fx;

    x = (float)j * 15.0f / 255.0f;
    x0 = (int)x;  fx = x - (float)x0;
    x1 = (x0 + 1 > 15) ? 15 : x0 + 1;
    for (int r = 0; r < 16; ++r)
      TC[(96 + r) * TC_STRIDE + j] = X2[r * 16 + x0] * (1.0f - fx) + X2[r * 16 + x1] * fx;
  }
  __syncthreads();

  // ---- maps = WY @ TC via V_WMMA_F32_16X16X4_F32, 16x16 tiles ----
  // A frag (16x4 f32, 2 VGPR): lanes 0-15 -> K=k4+0/+1, lanes 16-31 -> K=k4+2/+3
  // C/D frag (8 VGPR): vgpr r, lanes 0-15 row r, lanes 16-31 row r+8; col = lane&15
  const int lane    = tid & 31;
  const int wave    = tid >> 5;
  const int halfsel = lane >> 4;          // 0 | 1
  const int l15     = lane & 15;
  float* mp = maps + (size_t)b * (OUT_HW * OUT_HW);

  for (int ib = wave * 2; ib < wave * 2 + 2; ++ib) {   // 16 row-blocks / 8 waves
    const int row_a = ib * 16 + l15;
    for (int jb = 0; jb < 16; ++jb) {
      const int col_b = jb * 16 + l15;
      v8f acc{};
#pragma unroll 4
      for (int k4 = 0; k4 < K_TOT; k4 += 4) {
        const int kb = k4 + (halfsel << 1);
        v2f a, bm;
        a.x  = WY[row_a * WY_STRIDE + kb];
        a.y  = WY[row_a * WY_STRIDE + kb + 1];
        bm.x = TC[kb * TC_STRIDE + col_b];
        bm.y = TC[(kb + 1) * TC_STRIDE + col_b];
        acc = __builtin_amdgcn_wmma_f32_16x16x4_f32(
            false, a, false, bm, (short)0, acc, false, false);
      }
      const int rbase = ib * 16 + (halfsel << 3);
#pragma unroll
      for (int r = 0; r < 8; ++r)
        mp[(size_t)(rbase + r) * OUT_HW + jb * 16 + l15] = acc[r];
    }
  }
}

// ---- 5x5 reflect-pad gaussian blur (separable) + per-sample max --------------
__global__ void uninet_blur_max(const float* __restrict__ maps,
                                float* __restrict__ scores) {
  __shared__ float vrow[OUT_HW];
  __shared__ float red[256];
  const int c = threadIdx.x;              // thread owns one column
  const int b = blockIdx.x;
  const float* M = maps + (size_t)b * (OUT_HW * OUT_HW);

  float w[5]; float s = 0.0f;
#pragma unroll
  for (int k = 0; k < 5; ++k) {
    float d = (float)(k - 2);
    w[k] = expf(-(d * d) / 32.0f);        // sigma = 4
    s += w[k];
  }
#pragma unroll
  for (int k = 0; k < 5; ++k) w[k] /= s;

  // rolling 5-row window at y=0: rows refl(-2..2) = 2,1,0,1,2
  float m0 = M[2 * OUT_HW + c];
  float m1 = M[1 * OUT_HW + c];
  float m2 = M[0 * OUT_HW + c];
  float m3 = M[1 * OUT_HW + c];
  float m4 = M[2 * OUT_HW + c];

  // reflect column indices (constant per thread)
  int cm2 = c - 2; if (cm2 < 0) cm2 = -cm2;
  int cm1 = c - 1; if (cm1 < 0) cm1 = -cm1;
  int cp1 = c + 1; if (cp1 > 255) cp1 = 510 - cp1;
  int cp2 = c + 2; if (cp2 > 255) cp2 = 510 - cp2;

  float best = -INFINITY;
  for (int y = 0; y < OUT_HW; ++y) {
    float v = w[0] * m0 + w[1] * m1 + w[2] * m2 + w[3] * m3 + w[4] * m4;
    vrow[c] = v;
    __syncthreads();
    float h = w[0] * vrow[cm2] + w[1] * vrow[cm1] + w[2] * v +
              w[3] * vrow[cp1] + w[4] * vrow[cp2];
    best = fmaxf(best, h);
    __syncthreads();
    m0 = m1; m1 = m2; m2 = m3; m3 = m4;
    int yn = y + 3; if (yn > 255) yn = 510 - yn;   // reflect
    m4 = M[(size_t)yn * OUT_HW + c];
  }

  red[c] = best;
  __syncthreads();
  for (int st = 128; st > 0; st >>= 1) {
    if (c < st) red[c] = fmaxf(red[c], red[c + st]);
    __syncthreads();
  }
  if (c == 0) scores[b] = red[0];
}

extern "C" void kernel_launch(void* const* d_in, const int* in_sizes, int n_in,
                              void* d_out, int out_size, void* d_ws, size_t ws_size,
                              hipStream_t stream) {
  const float* in0 = (const float*)d_in[0];   // [B,64,64]
  const float* in1 = (const float*)d_in[1];   // [B,32,32]
  const float* in2 = (const float*)d_in[2];   // [B,16,16]
  float* out    = (float*)d_out;
  const int B   = in_sizes[0] / (64 * 64);
  float* scores = out;                        // [B,1]
  float* maps   = out + B;                    // [B,256,256]

  const size_t shmem = (size_t)SMEM_FLOATS * sizeof(float);
  uninet_upsample_gemm<<<dim3(B), dim3(256), shmem, stream>>>(in0, in1, in2, maps);
  uninet_blur_max<<<dim3(B), dim3(256), 0, stream>>>(maps, scores);
}